// GCN_24257975287859
// MI455X (gfx1250) — compile-verified
//
#include <hip/hip_runtime.h>
#include <math.h>

typedef __attribute__((ext_vector_type(2))) float v2f;
typedef __attribute__((ext_vector_type(8))) float v8f;

#define NN 100000
#define EE 1600000

// ---------------- normalization: deg -> dinv ----------------
__global__ void k_deg_init(float* __restrict__ deg) {
    int i = blockIdx.x * blockDim.x + threadIdx.x;
    if (i < NN) deg[i] = 1.0f;   // self-loop contribution
}

__global__ void k_deg_accum(const int* __restrict__ dst, float* __restrict__ deg) {
    int e = blockIdx.x * blockDim.x + threadIdx.x;
    if (e < EE) atomicAdd(&deg[dst[e]], 1.0f);
}

__global__ void k_rsqrt(float* __restrict__ dinv) {
    int i = blockIdx.x * blockDim.x + threadIdx.x;
    if (i < NN) {
        float d = dinv[i];
        dinv[i] = (d > 0.0f) ? rsqrtf(d) : 0.0f;
    }
}

// ---------------- fp32 WMMA GEMM: H[N,NC] = A[N,128] @ W[128,NC] ----------------
// One wave computes a 16-row tile across all NC columns (NC/16 accumulators),
// reusing the A fragment for every column tile. K is always 128.
template<int NC>
__global__ __launch_bounds__(256) void k_gemm(const float* __restrict__ A,
                                              const float* __restrict__ W,
                                              float* __restrict__ H) {
    constexpr int K  = 128;
    constexpr int NT = NC / 16;
    const int lane = threadIdx.x & 31;
    const int wave = threadIdx.x >> 5;
    const int rowTile = blockIdx.x * 8 + wave;      // wave-uniform
    if (rowTile * 16 >= NN) return;                 // whole wave exits together
    const int row0 = rowTile * 16;
    const int half = lane >> 4;                     // 0: K pair {0,1}; 1: {2,3}
    const int mn   = lane & 15;                     // M for A, N for B/C/D

    v8f acc[NT] = {};
    const float* arow = A + (size_t)(row0 + mn) * K + 2 * half;

    for (int k = 0; k < K; k += 4) {
        v2f a;
        a.x = arow[k];
        a.y = arow[k + 1];
#pragma unroll
        for (int c = 0; c < NT; ++c) {
            const float* bp = W + (size_t)(k + 2 * half) * NC + c * 16 + mn;
            v2f b;
            b.x = bp[0];
            b.y = bp[NC];
            acc[c] = __builtin_amdgcn_wmma_f32_16x16x4_f32(
                false, a, false, b, (short)0, acc[c], false, false);
        }
    }

#pragma unroll
    for (int c = 0; c < NT; ++c) {
#pragma unroll
        for (int r = 0; r < 8; ++r) {
            // C/D layout: VGPR r holds M=r (lanes 0-15) and M=8+r (lanes 16-31)
            H[(size_t)(row0 + half * 8 + r) * NC + c * 16 + mn] = acc[c][r];
        }
    }
}

// ---------------- self-loop init: out[i,c] = dinv[i]^2 * h[i,c] ----------------
template<int NC>
__global__ void k_self(const float* __restrict__ h, const float* __restrict__ dinv,
                       float* __restrict__ out) {
    int t = blockIdx.x * blockDim.x + threadIdx.x;
    if (t >= NN * NC) return;
    int i = t / NC;
    float di = dinv[i];
    out[t] = di * di * h[t];
}

// ---------------- edge scatter: out[dst] += dinv[s]*dinv[d] * h[src] ----------------
template<int NC>
__global__ void k_scatter(const float* __restrict__ h, const int* __restrict__ src,
                          const int* __restrict__ dst, const float* __restrict__ dinv,
                          float* __restrict__ out) {
    constexpr int G = NC / 4;                      // float4 groups per row
    int t = blockIdx.x * blockDim.x + threadIdx.x;
    if (t >= EE * G) return;
    int e = t / G;
    int g = t - e * G;
    int s = src[e];
    int d = dst[e];
    float nrm = dinv[s] * dinv[d];
    const float4 hv = *(const float4*)(h + (size_t)s * NC + g * 4);
    float* op = out + (size_t)d * NC + g * 4;
    atomicAdd(op + 0, nrm * hv.x);
    atomicAdd(op + 1, nrm * hv.y);
    atomicAdd(op + 2, nrm * hv.z);
    atomicAdd(op + 3, nrm * hv.w);
}

// ---------------- bias + relu ----------------
template<int NC>
__global__ void k_bias_relu(float* __restrict__ x, const float* __restrict__ b) {
    int t = blockIdx.x * blockDim.x + threadIdx.x;
    if (t >= NN * NC) return;
    int c = t & (NC - 1);
    float v = x[t] + b[c];
    x[t] = v > 0.0f ? v : 0.0f;
}

// ---------------- bias + log_softmax over 64 channels, one wave per row ----------------
__global__ __launch_bounds__(256) void k_bias_logsoftmax(float* __restrict__ x,
                                                         const float* __restrict__ b) {
    const int lane = threadIdx.x & 31;
    const int wave = threadIdx.x >> 5;
    int row = blockIdx.x * 8 + wave;
    if (row >= NN) return;
    float* p = x + (size_t)row * 64;
    float v0 = p[lane] + b[lane];
    float v1 = p[lane + 32] + b[lane + 32];
    float m = fmaxf(v0, v1);
#pragma unroll
    for (int off = 16; off >= 1; off >>= 1) m = fmaxf(m, __shfl_xor(m, off, 32));
    float s = expf(v0 - m) + expf(v1 - m);
#pragma unroll
    for (int off = 16; off >= 1; off >>= 1) s += __shfl_xor(s, off, 32);
    float lse = m + logf(s);
    p[lane]      = v0 - lse;
    p[lane + 32] = v1 - lse;
}

extern "C" void kernel_launch(void* const* d_in, const int* in_sizes, int n_in,
                              void* d_out, int out_size, void* d_ws, size_t ws_size,
                              hipStream_t stream) {
    const float* x  = (const float*)d_in[0];
    const int*   ei = (const int*)d_in[1];      // [2,E] row-major: src then dst
    const int*   src = ei;
    const int*   dst = ei + EE;
    const float* W1 = (const float*)d_in[2];
    const float* b1 = (const float*)d_in[3];
    const float* W2 = (const float*)d_in[4];
    const float* b2 = (const float*)d_in[5];
    const float* W3 = (const float*)d_in[6];
    const float* b3 = (const float*)d_in[7];
    float* out = (float*)d_out;

    char* ws = (char*)d_ws;
    float* dinv = (float*)ws;                                        // N floats (deg -> dinv)
    size_t off  = (((size_t)NN * 4) + 255) & ~(size_t)255;
    float* bufA = (float*)(ws + off);                                // N*128 floats
    float* bufB = bufA + (size_t)NN * 128;                           // N*128 floats

    const int T = 256;
    const int gN    = (NN + T - 1) / T;
    const int gE    = (EE + T - 1) / T;
    const int gNC128 = (NN * 128 + T - 1) / T;
    const int gNC64  = (NN * 64 + T - 1) / T;
    const int gSc128 = (EE * 32 + T - 1) / T;
    const int gSc64  = (EE * 16 + T - 1) / T;
    const int gGemm  = (NN / 16 + 7) / 8;
    const int gRow   = (NN + 7) / 8;

    // D^{-1/2}
    k_deg_init <<<gN, T, 0, stream>>>(dinv);
    k_deg_accum<<<gE, T, 0, stream>>>(dst, dinv);
    k_rsqrt    <<<gN, T, 0, stream>>>(dinv);

    // layer 1: h = x@W1 ; agg = D^-1/2 (A+I) D^-1/2 h ; relu(agg + b1)
    k_gemm<128>     <<<gGemm,  T, 0, stream>>>(x, W1, bufA);
    k_self<128>     <<<gNC128, T, 0, stream>>>(bufA, dinv, bufB);
    k_scatter<128>  <<<gSc128, T, 0, stream>>>(bufA, src, dst, dinv, bufB);
    k_bias_relu<128><<<gNC128, T, 0, stream>>>(bufB, b1);

    // layer 2
    k_gemm<128>     <<<gGemm,  T, 0, stream>>>(bufB, W2, bufA);
    k_self<128>     <<<gNC128, T, 0, stream>>>(bufA, dinv, bufB);
    k_scatter<128>  <<<gSc128, T, 0, stream>>>(bufA, src, dst, dinv, bufB);
    k_bias_relu<128><<<gNC128, T, 0, stream>>>(bufB, b2);

    // layer 3: accumulate straight into d_out, then bias + log_softmax in place
    k_gemm<64>      <<<gGemm,  T, 0, stream>>>(bufB, W3, bufA);
    k_self<64>      <<<gNC64,  T, 0, stream>>>(bufA, dinv, out);
    k_scatter<64>   <<<gSc64,  T, 0, stream>>>(bufA, src, dst, dinv, out);
    k_bias_logsoftmax<<<gRow,  T, 0, stream>>>(out, b3);
}